// SelectiveSSMBlock_33414845563025
// MI455X (gfx1250) — compile-verified
//
#include <hip/hip_runtime.h>
#include <math.h>

// ---------------- model constants (from reference) ----------------
#define L_SEQ     2048
#define DMODEL    1024
#define DINNER    2048
#define DSTATE    64
#define NHEADS    32
#define HEADDIM   64
#define CONVDIM   2176        // DINNER + 2*DSTATE
#define DINPROJ   4256        // 2*DINNER + 2*DSTATE + NHEADS
#define FFNHID    3072

typedef float v2f __attribute__((ext_vector_type(2)));
typedef float v8f __attribute__((ext_vector_type(8)));
typedef int   v4i __attribute__((ext_vector_type(4)));

// CDNA5 async global->LDS path (ASYNCcnt-tracked). The builtin takes typed
// b128 (int4) pointers: param0 = global (AS1), param1 = LDS (AS3), then
// immediate offset and cache-policy.
#if defined(__gfx1250__) && __has_builtin(__builtin_amdgcn_global_load_async_to_lds_b128)
#define HAVE_ASYNC_LDS 1
#define ASYNC_CP16(gptr, lptr)                                              \
  __builtin_amdgcn_global_load_async_to_lds_b128(                           \
      (__attribute__((address_space(1))) v4i*)(gptr),                       \
      (__attribute__((address_space(3))) v4i*)(lptr), 0, 0)
#else
#define HAVE_ASYNC_LDS 0
#endif

__device__ __forceinline__ void wait_async0() {
#if defined(__gfx1250__)
#if __has_builtin(__builtin_amdgcn_s_wait_asynccnt)
  __builtin_amdgcn_s_wait_asynccnt(0);
#else
  asm volatile("s_wait_asynccnt 0" ::: "memory");
#endif
#endif
}

__device__ __forceinline__ float siluf(float x) { return x / (1.0f + expf(-x)); }

__device__ __forceinline__ float block_reduce_sum(float v) {
#pragma unroll
  for (int off = 16; off > 0; off >>= 1) v += __shfl_xor(v, off, 32);
  __shared__ float tmp[8];
  const int wid = threadIdx.x >> 5, lane = threadIdx.x & 31;
  if (lane == 0) tmp[wid] = v;
  __syncthreads();
  if (wid == 0) {
    float r = (lane < 8) ? tmp[lane] : 0.0f;
#pragma unroll
    for (int off = 4; off > 0; off >>= 1) r += __shfl_xor(r, off, 32);
    if (lane == 0) tmp[0] = r;
  }
  __syncthreads();
  float out = tmp[0];
  __syncthreads();
  return out;
}

// ---------------- K1: residual mix + RMSNorm * ssm_norm_w ----------------
__global__ __launch_bounds__(256)
void k_mix_rms(const float* __restrict__ x, const float* __restrict__ x0,
               const float* __restrict__ mix, const float* __restrict__ nw,
               float* __restrict__ xr, float* __restrict__ u) {
  const int t = blockIdx.x;
  float v[4]; float ss = 0.0f;
#pragma unroll
  for (int i = 0; i < 4; ++i) {
    const int d = threadIdx.x + i * 256;
    const size_t idx = (size_t)t * DMODEL + d;
    float val = mix[d] * x[idx] + mix[DMODEL + d] * x0[idx];
    xr[idx] = val; v[i] = val; ss += val * val;
  }
  ss = block_reduce_sum(ss);
  const float rstd = rsqrtf(ss * (1.0f / DMODEL) + 1e-6f);
#pragma unroll
  for (int i = 0; i < 4; ++i) {
    const int d = threadIdx.x + i * 256;
    u[(size_t)t * DMODEL + d] = v[i] * rstd * nw[d];
  }
}

// ---------------- WMMA fp32 GEMM: C[M,N] = A[M,K] * W[N,K]^T ----------------
// 256 threads = 8 wave32; block tile 64x32, K-tile 32, double-buffered LDS.
// Async path: global->LDS DMA of tile k+1 overlaps the WMMA chain on tile k.
#define BM 64
#define BN 32
#define BK 32
#define LDP 36   // padded LDS row stride: float4-aligned, conflict-free column reads

__global__ __launch_bounds__(256)
void k_gemm_f32_wmma(float* __restrict__ C, const float* __restrict__ A,
                     const float* __restrict__ W, int M, int N, int K) {
  __shared__ float As[2][BM][LDP];
  __shared__ float Ws[2][BN][LDP];
  const int tid  = threadIdx.x;
  const int m0   = blockIdx.y * BM;
  const int n0   = blockIdx.x * BN;
  const int wid  = tid >> 5;
  const int lane = tid & 31;
  const int l16  = lane & 15;
  const int hi   = lane >> 4;           // 0: K pair {0,1}, 1: K pair {2,3}
  const int wm   = (wid >> 1) << 4;     // 0,16,32,48
  const int wn   = (wid & 1) << 4;      // 0,16
  const int ra   = tid >> 3;            // 0..31
  const int q4   = (tid & 7) << 2;      // 0..28 step 4

  const float* gA0 = &A[(size_t)(m0 + ra)      * K + q4];
  const float* gA1 = &A[(size_t)(m0 + ra + 32) * K + q4];
  const float* gW  = &W[(size_t)(n0 + ra)      * K + q4];

  v8f acc = {0.f, 0.f, 0.f, 0.f, 0.f, 0.f, 0.f, 0.f};

  // ---- prologue: stage K-tile 0 into buffer 0 ----
#if HAVE_ASYNC_LDS
  ASYNC_CP16(gA0, &As[0][ra][q4]);
  ASYNC_CP16(gA1, &As[0][ra + 32][q4]);
  ASYNC_CP16(gW,  &Ws[0][ra][q4]);
  wait_async0();
#else
  {
    float4 a0 = *(const float4*)gA0;
    float4 a1 = *(const float4*)gA1;
    float4 w0 = *(const float4*)gW;
    *(float4*)&As[0][ra][q4]      = a0;
    *(float4*)&As[0][ra + 32][q4] = a1;
    *(float4*)&Ws[0][ra][q4]      = w0;
  }
#endif
  __syncthreads();

  int buf = 0;
  for (int kt = 0; kt < K; kt += BK) {
    const bool has_next = (kt + BK) < K;
    const int nb = buf ^ 1;

    // ---- issue staging of tile kt+BK into the alternate buffer ----
#if HAVE_ASYNC_LDS
    if (has_next) {
      ASYNC_CP16(gA0 + kt + BK, &As[nb][ra][q4]);
      ASYNC_CP16(gA1 + kt + BK, &As[nb][ra + 32][q4]);
      ASYNC_CP16(gW  + kt + BK, &Ws[nb][ra][q4]);
    }
#else
    float4 na0, na1, nw0;
    if (has_next) {                    // loads in flight during compute
      na0 = *(const float4*)(gA0 + kt + BK);
      na1 = *(const float4*)(gA1 + kt + BK);
      nw0 = *(const float4*)(gW  + kt + BK);
      __builtin_prefetch(gA0 + kt + 2 * BK, 0, 3);
      __builtin_prefetch(gW  + kt + 2 * BK, 0, 3);
    }
#endif

    // ---- compute: 8 chained v_wmma_f32_16x16x4_f32 on current buffer ----
#pragma unroll
    for (int kk = 0; kk < BK; kk += 4) {
      const int k = kk + (hi << 1);
      v2f a = *(const v2f*)&As[buf][wm + l16][k];  // A frag: row=l16, K pair by half-wave
      v2f b = *(const v2f*)&Ws[buf][wn + l16][k];  // B frag: col=l16 (B[k][n]=W[n][k])
      acc = __builtin_amdgcn_wmma_f32_16x16x4_f32(
          false, a, false, b, (short)0, acc, false, false);
    }

    // ---- complete staging + publish ----
#if HAVE_ASYNC_LDS
    if (has_next) wait_async0();
#else
    if (has_next) {
      *(float4*)&As[nb][ra][q4]      = na0;
      *(float4*)&As[nb][ra + 32][q4] = na1;
      *(float4*)&Ws[nb][ra][q4]      = nw0;
    }
#endif
    __syncthreads();
    buf = nb;
  }

  // D layout: VGPR j -> M = j (lanes 0-15) / j+8 (lanes 16-31), N = l16
  const int row = m0 + wm + (hi << 3);
  const int col = n0 + wn + l16;
#pragma unroll
  for (int j = 0; j < 8; ++j)
    C[(size_t)(row + j) * N + col] = acc[j];
}

// ---------------- K3: depthwise causal conv(4) + SiLU ----------------
__global__ __launch_bounds__(256)
void k_conv_silu(const float* __restrict__ zx, const float* __restrict__ cw,
                 const float* __restrict__ cb, float* __restrict__ xc) {
  const int t = blockIdx.x;
  for (int c = threadIdx.x; c < CONVDIM; c += 256) {
    float acc = cb[c];
#pragma unroll
    for (int k = 0; k < 4; ++k) {
      const int tt = t - 3 + k;
      if (tt >= 0)
        acc += zx[(size_t)tt * DINPROJ + DINNER + c] * cw[c * 4 + k];
    }
    xc[(size_t)t * CONVDIM + c] = siluf(acc);
  }
}

// ---------------- K4: dt softplus + dA = exp(-exp(A_log)*dt) ----------------
__global__ __launch_bounds__(256)
void k_dt(const float* __restrict__ zx, const float* __restrict__ dtb,
          const float* __restrict__ alog, float* __restrict__ dts,
          float* __restrict__ dAs) {
  const int idx = blockIdx.x * 256 + threadIdx.x;   // t*NHEADS + h
  const int t = idx / NHEADS, h = idx % NHEADS;
  const float raw = zx[(size_t)t * DINPROJ + (DINNER + CONVDIM) + h] + dtb[h];
  const float sp  = (raw > 20.0f) ? raw : log1pf(expf(raw));
  dts[idx] = sp;
  dAs[idx] = expf(-expf(alog[h]) * sp);
}

// ---------------- K5: selective scan, one workgroup per head ----------------
// state h[p][n] lives in registers: lane = p*4 + ng, each lane owns 16 n's.
__global__ __launch_bounds__(256)
void k_scan(const float* __restrict__ xc, const float* __restrict__ dts,
            const float* __restrict__ dAs, const float* __restrict__ Dp,
            float* __restrict__ y) {
  const int head = blockIdx.x;
  const int p  = threadIdx.x >> 2;
  const int ng = threadIdx.x & 3;
  float h[16];
#pragma unroll
  for (int i = 0; i < 16; ++i) h[i] = 0.0f;
  const float dparam = Dp[head];
  const float* row = xc;
  for (int t = 0; t < L_SEQ; ++t, row += CONVDIM) {
    const float da  = dAs[t * NHEADS + head];
    const float dtv = dts[t * NHEADS + head];
    const float xp  = row[head * HEADDIM + p];
    const float4* Bp = (const float4*)(row + DINNER + ng * 16);
    const float4* Cp = (const float4*)(row + DINNER + DSTATE + ng * 16);
    float4 b4[4], c4[4];
#pragma unroll
    for (int i = 0; i < 4; ++i) { b4[i] = Bp[i]; c4[i] = Cp[i]; }
    if (t + 1 < L_SEQ) {  // keep next timestep's rows near (L2/WGP)
      __builtin_prefetch(row + CONVDIM + head * HEADDIM + p, 0, 3);
      __builtin_prefetch(row + CONVDIM + DINNER + ng * 16, 0, 3);
    }
    const float s = dtv * xp;
    float pa[4] = {0.f, 0.f, 0.f, 0.f};   // 4 partials: short dependency chains
    const float* bb = (const float*)b4;
    const float* cc = (const float*)c4;
#pragma unroll
    for (int i = 0; i < 16; ++i) {
      h[i] = fmaf(da, h[i], s * bb[i]);
      pa[i & 3] = fmaf(h[i], cc[i], pa[i & 3]);
    }
    float acc = (pa[0] + pa[1]) + (pa[2] + pa[3]);
    acc += __shfl_xor(acc, 1, 32);
    acc += __shfl_xor(acc, 2, 32);
    if (ng == 0)
      y[(size_t)t * DINNER + head * HEADDIM + p] = acc + dparam * xp;
  }
}

// ---------------- K6: y = rms(y * silu(z)) * mnorm_w (in-place) ----------------
__global__ __launch_bounds__(256)
void k_gate_rms(float* __restrict__ y, const float* __restrict__ zx,
                const float* __restrict__ mnorm) {
  const int t = blockIdx.x;
  float g[8]; float ss = 0.0f;
#pragma unroll
  for (int i = 0; i < 8; ++i) {
    const int d = threadIdx.x + i * 256;
    const float z = zx[(size_t)t * DINPROJ + d];
    const float v = y[(size_t)t * DINNER + d] * siluf(z);
    g[i] = v; ss += v * v;
  }
  ss = block_reduce_sum(ss);
  const float rstd = rsqrtf(ss * (1.0f / DINNER) + 1e-6f);
#pragma unroll
  for (int i = 0; i < 8; ++i) {
    const int d = threadIdx.x + i * 256;
    y[(size_t)t * DINNER + d] = g[i] * rstd * mnorm[d];
  }
}

// ---------------- K8: x2 = xr + ssm_scale*y2 (in place); h = rms(x2)*ffn_norm ----
__global__ __launch_bounds__(256)
void k_resid_rms(float* __restrict__ xr, const float* __restrict__ y2,
                 const float* __restrict__ scale, const float* __restrict__ nw,
                 float* __restrict__ hb) {
  const int t = blockIdx.x;
  float v[4]; float ss = 0.0f;
#pragma unroll
  for (int i = 0; i < 4; ++i) {
    const int d = threadIdx.x + i * 256;
    const size_t idx = (size_t)t * DMODEL + d;
    const float val = xr[idx] + scale[d] * y2[idx];
    xr[idx] = val; v[i] = val; ss += val * val;
  }
  ss = block_reduce_sum(ss);
  const float rstd = rsqrtf(ss * (1.0f / DMODEL) + 1e-6f);
#pragma unroll
  for (int i = 0; i < 4; ++i) {
    const int d = threadIdx.x + i * 256;
    hb[(size_t)t * DMODEL + d] = v[i] * rstd * nw[d];
  }
}

// ---------------- K10: SwiGLU ----------------
__global__ __launch_bounds__(256)
void k_swiglu(const float* __restrict__ gu, float* __restrict__ act) {
  const size_t idx = (size_t)blockIdx.x * 256 + threadIdx.x;  // L_SEQ*FFNHID
  const size_t t = idx / FFNHID, c = idx % FFNHID;
  const float g  = gu[t * (2 * FFNHID) + c];
  const float up = gu[t * (2 * FFNHID) + FFNHID + c];
  act[idx] = siluf(g) * up;
}

// ---------------- K12: out = x2 + mlp_scale * down ----------------
__global__ __launch_bounds__(256)
void k_final(const float* __restrict__ x2, const float* __restrict__ dn,
             const float* __restrict__ mscale, float* __restrict__ out) {
  const size_t idx = (size_t)blockIdx.x * 256 + threadIdx.x;
  const int d = (int)(idx % DMODEL);
  out[idx] = x2[idx] + mscale[d] * dn[idx];
}

// ---------------- host-side launch ----------------
extern "C" void kernel_launch(void* const* d_in, const int* in_sizes, int n_in,
                              void* d_out, int out_size, void* d_ws, size_t ws_size,
                              hipStream_t stream) {
  (void)in_sizes; (void)n_in; (void)out_size; (void)ws_size;
  const float* x          = (const float*)d_in[0];
  const float* x0         = (const float*)d_in[1];
  /* d_in[2] = mem: unused by the reference */
  const float* resid_mix  = (const float*)d_in[3];
  const float* ssm_scale  = (const float*)d_in[4];
  const float* mlp_scale  = (const float*)d_in[5];
  const float* ssm_norm_w = (const float*)d_in[6];
  const float* ffn_norm_w = (const float*)d_in[7];
  const float* in_proj_w  = (const float*)d_in[8];
  const float* conv_w     = (const float*)d_in[9];
  const float* conv_b     = (const float*)d_in[10];
  const float* dt_bias    = (const float*)d_in[11];
  const float* A_log      = (const float*)d_in[12];
  const float* D_param    = (const float*)d_in[13];
  const float* mnorm_w    = (const float*)d_in[14];
  const float* out_proj_w = (const float*)d_in[15];
  const float* gate_up_w  = (const float*)d_in[16];
  const float* down_w     = (const float*)d_in[17];
  float* out = (float*)d_out;
  float* ws  = (float*)d_ws;

  // workspace layout (floats), with lifetime-safe reuse (~112 MB total):
  float* xr  = ws;                                   // 2048*1024 ; -> x2 in place
  float* u   = xr + (size_t)L_SEQ * DMODEL;          // 2048*1024 ; -> y2 -> down-out
  float* big = u  + (size_t)L_SEQ * DMODEL;
  float* zx  = big;                                  // 2048*4256 (zxbcdt)
  float* xc  = zx + (size_t)L_SEQ * DINPROJ;         // 2048*2176 (conv output)
  float* gu  = big;                                  // 2048*6144, reuses zx+xc (dead)
  float* dts = xc + (size_t)L_SEQ * CONVDIM;         // 2048*32
  float* dAs = dts + (size_t)L_SEQ * NHEADS;         // 2048*32
  float* y   = dAs + (size_t)L_SEQ * NHEADS;         // 2048*2048 ; -> ssm_in in place
  float* act = y  + (size_t)L_SEQ * DINNER;          // 2048*3072
  float* hb  = act;                                  // 2048*1024 (dead before act)

  const dim3 blk(256);
  k_mix_rms<<<L_SEQ, blk, 0, stream>>>(x, x0, resid_mix, ssm_norm_w, xr, u);
  k_gemm_f32_wmma<<<dim3(DINPROJ / BN, L_SEQ / BM), blk, 0, stream>>>(
      zx, u, in_proj_w, L_SEQ, DINPROJ, DMODEL);
  k_conv_silu<<<L_SEQ, blk, 0, stream>>>(zx, conv_w, conv_b, xc);
  k_dt<<<(L_SEQ * NHEADS) / 256, blk, 0, stream>>>(zx, dt_bias, A_log, dts, dAs);
  k_scan<<<NHEADS, blk, 0, stream>>>(xc, dts, dAs, D_param, y);
  k_gate_rms<<<L_SEQ, blk, 0, stream>>>(y, zx, mnorm_w);
  k_gemm_f32_wmma<<<dim3(DMODEL / BN, L_SEQ / BM), blk, 0, stream>>>(
      u, y, out_proj_w, L_SEQ, DMODEL, DINNER);
  k_resid_rms<<<L_SEQ, blk, 0, stream>>>(xr, u, ssm_scale, ffn_norm_w, hb);
  k_gemm_f32_wmma<<<dim3((2 * FFNHID) / BN, L_SEQ / BM), blk, 0, stream>>>(
      gu, hb, gate_up_w, L_SEQ, 2 * FFNHID, DMODEL);
  k_swiglu<<<(L_SEQ * FFNHID) / 256, blk, 0, stream>>>(gu, act);
  k_gemm_f32_wmma<<<dim3(DMODEL / BN, L_SEQ / BM), blk, 0, stream>>>(
      u, act, down_w, L_SEQ, DMODEL, FFNHID);
  k_final<<<(L_SEQ * DMODEL) / 256, blk, 0, stream>>>(xr, u, mlp_scale, out);
}